// GCN_481036337415
// MI455X (gfx1250) — compile-verified
//
#include <hip/hip_runtime.h>
#include <hip/hip_bf16.h>

typedef __attribute__((ext_vector_type(2))) float v2f;
typedef __attribute__((ext_vector_type(8))) float v8f;

#define NODES_PER_TILE 16

// ---------------- utility ----------------
__global__ void fill_zero_f32(float* __restrict__ p, long n) {
  long i = (long)blockIdx.x * blockDim.x + threadIdx.x;
  if (i < n) p[i] = 0.0f;
}

// ---------------- degree / norm ----------------
__global__ void deg_kernel(const int* __restrict__ dst, float* __restrict__ deg, int ne) {
  int e = blockIdx.x * blockDim.x + threadIdx.x;
  if (e >= ne) return;
  atomicAdd(&deg[dst[e]], 1.0f);
}

__global__ void dinv_kernel(float* __restrict__ deg, int n) {
  int i = blockIdx.x * blockDim.x + threadIdx.x;
  if (i >= n) return;
  deg[i] = rsqrtf(deg[i] + 1.0f);   // in place: deg buffer becomes dinv
}

// ---------------- h @ W.T (tiny dims, fully unrolled) ----------------
template <int IN, int OUT>
__global__ void xw_kernel(const float* __restrict__ h, const float* __restrict__ W,
                          float* __restrict__ tmp, int n) {
  int i = blockIdx.x * blockDim.x + threadIdx.x;
  if (i >= n) return;
  float hv[IN];
#pragma unroll
  for (int k = 0; k < IN; ++k) hv[k] = h[(long)i * IN + k];
#pragma unroll
  for (int o = 0; o < OUT; ++o) {
    float s = 0.0f;
#pragma unroll
    for (int k = 0; k < IN; ++k) s += hv[k] * W[o * IN + k];
    tmp[(long)i * OUT + o] = s;
  }
}

// ---------------- edge aggregation: agg[dst] += dinv[src]*dinv[dst]*tmp[src] ----------------
template <int OUT>
__global__ void edge_kernel(const int* __restrict__ src, const int* __restrict__ dst,
                            const float* __restrict__ dinv, const float* __restrict__ tmp,
                            float* __restrict__ agg, int ne) {
  int e = blockIdx.x * blockDim.x + threadIdx.x;
  if (e >= ne) return;
  int s = src[e];
  int d = dst[e];
  float w = dinv[s] * dinv[d];
#pragma unroll
  for (int o = 0; o < OUT; ++o)
    atomicAdd(&agg[(long)d * OUT + o], w * tmp[(long)s * OUT + o]);
}

// ---------------- combine: out = act(agg + dinv^2 * tmp + b) ----------------
// ACT: 0 = relu, 1 = tanh
template <int OUT, int ACT>
__global__ void combine_kernel(const float* __restrict__ agg, const float* __restrict__ tmp,
                               const float* __restrict__ dinv, const float* __restrict__ b,
                               float* __restrict__ out, int n) {
  int i = blockIdx.x * blockDim.x + threadIdx.x;
  if (i >= n) return;
  float sn = dinv[i] * dinv[i];
#pragma unroll
  for (int o = 0; o < OUT; ++o) {
    float v = agg[(long)i * OUT + o] + sn * tmp[(long)i * OUT + o] + b[o];
    if (ACT == 0) v = v > 0.0f ? v : 0.0f;
    else          v = tanhf(v);
    out[(long)i * OUT + o] = v;
  }
}

// ---------------- classifier: out[16x112 per node block] via V_WMMA_F32_16X16X4_F32 ----------------
// A (16x4 f32): lanes 0-15 carry K=0,1 in a[0],a[1]; lanes 16-31 carry K=2,3 (zero pad since feat dim=2)
// B (4x16 f32): VGPR0 lanes 0-15 = row K=0 (N=lane), lanes 16-31 = row K=2; VGPR1 = K=1 / K=3
// D (16x16 f32): VGPR r, lanes 0-15 -> M=r, N=lane; lanes 16-31 -> M=r+8, N=lane-16
__global__ void classifier_wmma_kernel(const float* __restrict__ h,   // [n, 2]
                                       const float* __restrict__ Wc,  // [112, 2] row-major
                                       const float* __restrict__ bc,  // [112]
                                       float* __restrict__ out,       // [n, 112]
                                       int n) {
  int nodeBlk = blockIdx.x;           // n/16 blocks (n divisible by 16)
  int tile    = blockIdx.y;           // 7 tiles of 16 output columns
  int lane    = threadIdx.x;          // wave32: 0..31
  int half    = lane >> 4;            // 0: K=0,1 ; 1: K=2,3 (zero pad)
  int l       = lane & 15;

  v2f a = {0.0f, 0.0f};
  v2f b = {0.0f, 0.0f};
  if (half == 0) {
    int node = nodeBlk * NODES_PER_TILE + l;
    a.x = h[(long)node * 2 + 0];
    a.y = h[(long)node * 2 + 1];
    int col = tile * 16 + l;
    b.x = Wc[col * 2 + 0];     // Wc.T[k=0][col]
    b.y = Wc[col * 2 + 1];     // Wc.T[k=1][col]
  }

  v8f c = {};
  c = __builtin_amdgcn_wmma_f32_16x16x4_f32(
      /*neg_a=*/false, a, /*neg_b=*/false, b,
      /*c_mod=*/(short)0, c, /*reuse_a=*/false, /*reuse_b=*/false);

  int col = tile * 16 + l;
  float bias = bc[col];
#pragma unroll
  for (int r = 0; r < 8; ++r) {
    int m = r + half * 8;
    int node = nodeBlk * NODES_PER_TILE + m;
    out[(long)node * 112 + col] = c[r] + bias;
  }
}

// ---------------- launch ----------------
extern "C" void kernel_launch(void* const* d_in, const int* in_sizes, int n_in,
                              void* d_out, int out_size, void* d_ws, size_t ws_size,
                              hipStream_t stream) {
  const float* x   = (const float*)d_in[0];
  const int*   ei  = (const int*)d_in[1];
  const float* W1  = (const float*)d_in[2];
  const float* b1  = (const float*)d_in[3];
  const float* W2  = (const float*)d_in[4];
  const float* b2  = (const float*)d_in[5];
  const float* W3  = (const float*)d_in[6];
  const float* b3  = (const float*)d_in[7];
  const float* W4  = (const float*)d_in[8];
  const float* b4  = (const float*)d_in[9];
  const float* Wc  = (const float*)d_in[10];
  const float* bc  = (const float*)d_in[11];
  float* out = (float*)d_out;

  const int n  = in_sizes[0] / 8;    // 500000
  const int ne = in_sizes[1] / 2;    // 16000000
  const int* src = ei;               // edge_index[0]
  const int* dst = ei + ne;          // edge_index[1]

  // workspace layout (floats)
  float* wsf  = (float*)d_ws;
  float* dinv = wsf;                 // [n]      (deg -> dinv in place)
  float* bufT = dinv + n;            // [n*4]    h @ W.T
  float* bufA = bufT + (long)n * 4;  // [n*4]    aggregation accumulator
  float* bufH = bufA + (long)n * 4;  // [n*4]    activated hidden state

  const int B = 256;
  const int gN  = (n  + B - 1) / B;
  const int gN4 = ((n * 4) + B - 1) / B;
  const int gE  = (ne + B - 1) / B;

  float* hOut = out + (long)n * 112; // final h lives in second output section

  // degrees -> dinv
  fill_zero_f32<<<gN, B, 0, stream>>>(dinv, n);
  deg_kernel<<<gE, B, 0, stream>>>(dst, dinv, ne);
  dinv_kernel<<<gN, B, 0, stream>>>(dinv, n);

  // Layer 1: 8 -> 4, relu
  xw_kernel<8, 4><<<gN, B, 0, stream>>>(x, W1, bufT, n);
  fill_zero_f32<<<gN4, B, 0, stream>>>(bufA, (long)n * 4);
  edge_kernel<4><<<gE, B, 0, stream>>>(src, dst, dinv, bufT, bufA, ne);
  combine_kernel<4, 0><<<gN, B, 0, stream>>>(bufA, bufT, dinv, b1, bufH, n);

  // Layer 2: 4 -> 4, tanh
  xw_kernel<4, 4><<<gN, B, 0, stream>>>(bufH, W2, bufT, n);
  fill_zero_f32<<<gN4, B, 0, stream>>>(bufA, (long)n * 4);
  edge_kernel<4><<<gE, B, 0, stream>>>(src, dst, dinv, bufT, bufA, ne);
  combine_kernel<4, 1><<<gN, B, 0, stream>>>(bufA, bufT, dinv, b2, bufH, n);

  // Layer 3: 4 -> 2, relu
  xw_kernel<4, 2><<<gN, B, 0, stream>>>(bufH, W3, bufT, n);
  fill_zero_f32<<<gN4, B, 0, stream>>>(bufA, (long)n * 2);
  edge_kernel<2><<<gE, B, 0, stream>>>(src, dst, dinv, bufT, bufA, ne);
  combine_kernel<2, 0><<<gN, B, 0, stream>>>(bufA, bufT, dinv, b3, bufH, n);

  // Layer 4: 2 -> 2, tanh -> directly into d_out's h section
  xw_kernel<2, 2><<<gN, B, 0, stream>>>(bufH, W4, bufT, n);
  fill_zero_f32<<<gN4, B, 0, stream>>>(bufA, (long)n * 2);
  edge_kernel<2><<<gE, B, 0, stream>>>(src, dst, dinv, bufT, bufA, ne);
  combine_kernel<2, 1><<<gN, B, 0, stream>>>(bufA, bufT, dinv, b4, hOut, n);

  // Classifier: h[n,2] @ Wc.T -> out[n,112], WMMA f32 16x16x4
  dim3 grid(n / NODES_PER_TILE, 7);
  classifier_wmma_kernel<<<grid, 32, 0, stream>>>(hOut, Wc, bc, out, n);
}